// GraphNN_38723425141000
// MI455X (gfx1250) — compile-verified
//
#include <hip/hip_runtime.h>
#include <math.h>

#define NN 4096
#define CUTOFF 3.6f
#define CHUNK 256          // panel rows per TDM chunk (16 KB)
#define TPW 4              // row-tiles per wave

typedef float v2f __attribute__((ext_vector_type(2)));
typedef float v8f __attribute__((ext_vector_type(8)));
typedef unsigned int v4u __attribute__((ext_vector_type(4)));
typedef int v4i __attribute__((ext_vector_type(4)));
typedef int v8i __attribute__((ext_vector_type(8)));

// workspace layout (float offsets)
#define WS_PX    0
#define WS_PY    (WS_PX + NN)
#define WS_PZ    (WS_PY + NN)
#define WS_FEATS (WS_PZ + NN)            // NN*7
#define WS_S     (WS_FEATS + NN*7)       // NN
#define WS_PART  (WS_S + NN)             // 64
#define WS_SMAX  (WS_PART + 64)          // 1 (padded)
#define WS_F16   (WS_SMAX + 64)          // NN*16 padded panel: [e*f0..e*f6, e, px,py,pz, 0...]
#define WS_G     (WS_F16 + NN*16)        // NN*8

// ---------------- Kernel 1: per-row MLP -> feats, s, packed positions; block max of s
__global__ void __launch_bounds__(256) k_mlp(const float* __restrict__ x,
                                             const float* __restrict__ w1, const float* __restrict__ b1,
                                             const float* __restrict__ w2, const float* __restrict__ b2,
                                             const float* __restrict__ w3, const float* __restrict__ b3,
                                             float* __restrict__ ws) {
    int i = blockIdx.x * blockDim.x + threadIdx.x;
    __shared__ float red[256];

    float xi[7];
#pragma unroll
    for (int d = 0; d < 7; ++d) xi[d] = x[i * 7 + d];
    ws[WS_PX + i] = xi[0];
    ws[WS_PY + i] = xi[1];
    ws[WS_PZ + i] = xi[2];

    float h1[8];
#pragma unroll
    for (int h = 0; h < 8; ++h) {
        float acc = b1[h];
#pragma unroll
        for (int d = 0; d < 7; ++d) acc += xi[d] * w1[d * 8 + h];
        h1[h] = atanf(acc);
    }
    float h2[8];
#pragma unroll
    for (int h = 0; h < 8; ++h) {
        float acc = b2[h];
#pragma unroll
        for (int k = 0; k < 8; ++k) acc += h1[k] * w2[k * 8 + h];
        h2[h] = atanf(acc);
    }
    float m[23];
#pragma unroll
    for (int c = 0; c < 23; ++c) {
        float acc = b3[c];
#pragma unroll
        for (int k = 0; k < 8; ++k) acc += h2[k] * w3[k * 23 + c];
        m[c] = acc;
    }
#pragma unroll
    for (int d = 0; d < 7; ++d) ws[WS_FEATS + i * 7 + d] = m[d];

    float s = 0.f;
#pragma unroll
    for (int k = 0; k < 8; ++k) s += m[15 + k] * m[7 + k];  // queries . keys
    ws[WS_S + i] = s;

    red[threadIdx.x] = s;
    __syncthreads();
    for (int off = 128; off > 0; off >>= 1) {
        if (threadIdx.x < (unsigned)off) red[threadIdx.x] = fmaxf(red[threadIdx.x], red[threadIdx.x + off]);
        __syncthreads();
    }
    if (threadIdx.x == 0) ws[WS_PART + blockIdx.x] = red[0];
}

// ---------------- Kernel 1b: finish global max
__global__ void k_maxred(float* __restrict__ ws, int nblocks) {
    if (threadIdx.x == 0) {
        float mx = ws[WS_PART];
        for (int b = 1; b < nblocks; ++b) mx = fmaxf(mx, ws[WS_PART + b]);
        ws[WS_SMAX] = mx;
    }
}

// ---------------- Kernel 2: build padded panel row j = [e*f0..e*f6, e, px, py, pz, 0,0,0,0,0]
__global__ void __launch_bounds__(256) k_buildF(float* __restrict__ ws) {
    int j = blockIdx.x * blockDim.x + threadIdx.x;
    float e = expf(ws[WS_S + j] - ws[WS_SMAX]);
    float* Frow = ws + WS_F16 + j * 16;
#pragma unroll
    for (int d = 0; d < 7; ++d) Frow[d] = ws[WS_FEATS + j * 7 + d] * e;
    Frow[7]  = e;
    Frow[8]  = ws[WS_PX + j];   // positions ride in B columns 8..10 (D cols 8..15 are discarded)
    Frow[9]  = ws[WS_PY + j];
    Frow[10] = ws[WS_PZ + j];
#pragma unroll
    for (int d = 11; d < 16; ++d) Frow[d] = 0.f;
}

// ---------------- TDM: issue a 1-D 16KB panel-chunk DMA Global->LDS (D# per CDNA5 ISA)
__device__ __forceinline__ void tdm_issue_chunk(const float* __restrict__ gsrc, unsigned lds_byte_off) {
    unsigned long long ga = (unsigned long long)(uintptr_t)gsrc;
    v4u g0 = {
        1u,                                               // count=1, is_restore=0, gather=0
        lds_byte_off,                                     // lds_addr
        (unsigned)ga,                                     // global_addr[31:0]
        (unsigned)((ga >> 32) & 0x01FFFFFFu) | (2u << 30) // global_addr[56:32], type=2 (image)
    };
    v8i g1 = {
        (int)(2u << 16),              // wg_mask=0, data_size=2 (4B), no barrier/iterate/pad
        (int)((CHUNK * 16u) << 16),   // atomic_barrier_addr=0 | tensor_dim0[15:0]=4096
        (int)(1u << 16),              // tensor_dim0[31:16]=0 | tensor_dim1[15:0]=1
        (int)((CHUNK * 16u) << 16),   // tensor_dim1[31:16]=0 | tile_dim0=4096
        1,                            // tile_dim1=1 | tile_dim2=0
        (int)(CHUNK * 16u),           // tensor_dim0_stride[31:0]
        0,                            // stride0[47:32] | stride1[15:0]
        0                             // stride1[47:16]
    };
    v4i z4 = {0, 0, 0, 0};
    v8i z8 = {0, 0, 0, 0, 0, 0, 0, 0};
    // clang-23 / therock-10.0 form: (g0 v4u, g1 v8i, g2 v4i, g3 v4i, v8i, i32 cpol)
    __builtin_amdgcn_tensor_load_to_lds(g0, g1, z4, z4, z8, 0);
}

// ---------------- Kernel 3: G = A @ F via V_WMMA_F32_16X16X4_F32
// 8 waves/block, 4 row-tiles per wave, panel double-buffered through LDS by the Tensor Data Mover
__global__ void __launch_bounds__(256) k_adjmm(float* __restrict__ ws) {
    __shared__ float sbuf[2 * CHUNK * 16];  // 32 KB double buffer
    const float* __restrict__ F16g = ws + WS_F16;

    const int wave = threadIdx.x >> 5;
    const int lane = threadIdx.x & 31;
    const int tbase = (blockIdx.x * 8 + wave) * TPW;   // first of TPW row-tiles
    const int koff = (lane >> 4) * 2;                  // K pair owned by this half-wave
    const int ncol = lane & 15;                        // B/D column (N)

    float pix[TPW], piy[TPW], piz[TPW];
#pragma unroll
    for (int t = 0; t < TPW; ++t) {
        int mrow = (tbase + t) * 16 + (lane & 15);
        pix[t] = ws[WS_PX + mrow];
        piy[t] = ws[WS_PY + mrow];
        piz[t] = ws[WS_PZ + mrow];
    }

    v8f acc[TPW];
#pragma unroll
    for (int t = 0; t < TPW; ++t) acc[t] = (v8f){0.f, 0.f, 0.f, 0.f, 0.f, 0.f, 0.f, 0.f};

    const unsigned lds_base = (unsigned)(uintptr_t)(void*)&sbuf[0];

    if (wave == 0) {
        tdm_issue_chunk(F16g, lds_base);                 // preload chunk 0 -> buffer 0
        __builtin_amdgcn_s_wait_tensorcnt(0);
    }
    __syncthreads();

    for (int c = 0; c < NN / CHUNK; ++c) {
        if (wave == 0 && (c + 1) < NN / CHUNK) {         // prefetch next chunk into other buffer
            tdm_issue_chunk(F16g + (c + 1) * CHUNK * 16,
                            lds_base + ((c + 1) & 1) * (CHUNK * 16 * 4));
        }
        const float* __restrict__ buf = sbuf + (c & 1) * (CHUNK * 16);

#pragma unroll 2
        for (int jl = 0; jl < CHUNK; jl += 4) {
            const float* __restrict__ r0 = buf + (jl + koff) * 16;
            const float* __restrict__ r1 = r0 + 16;
            float bx0 = r0[8], by0 = r0[9], bz0 = r0[10];
            float bx1 = r1[8], by1 = r1[9], bz1 = r1[10];
            v2f B = {r0[ncol], r1[ncol]};
#pragma unroll
            for (int t = 0; t < TPW; ++t) {
                float a0 = (fabsf(pix[t] - bx0) + fabsf(piy[t] - by0) + fabsf(piz[t] - bz0)) <= CUTOFF ? 1.0f : 0.0f;
                float a1 = (fabsf(pix[t] - bx1) + fabsf(piy[t] - by1) + fabsf(piz[t] - bz1)) <= CUTOFF ? 1.0f : 0.0f;
                v2f A = {a0, a1};
                acc[t] = __builtin_amdgcn_wmma_f32_16x16x4_f32(false, A, false, B, (short)0, acc[t], false, false);
            }
        }

        __syncthreads();                                 // everyone done reading buf (and old other-buf)
        if (wave == 0) __builtin_amdgcn_s_wait_tensorcnt(0);
        __syncthreads();                                 // next chunk visible to all waves
    }

    if (ncol < 8) {
        float* __restrict__ G = ws + WS_G;
#pragma unroll
        for (int t = 0; t < TPW; ++t) {
            int rbase = (tbase + t) * 16 + (lane >> 4) * 8;
#pragma unroll
            for (int r = 0; r < 8; ++r) G[(rbase + r) * 8 + ncol] = acc[t][r];
        }
    }
}

// ---------------- Kernel 4: inclusive prefix sum down each of the 8 G columns (in place)
__global__ void k_scan(float* __restrict__ ws) {
    int c = threadIdx.x;
    if (c < 8) {
        float* __restrict__ G = ws + WS_G;
        float acc = 0.f;
        for (int i = 0; i < NN; ++i) {
            acc += G[i * 8 + c];
            G[i * 8 + c] = acc;
        }
    }
}

// ---------------- Kernel 5: agg = P[:,0:7]/P[:,7]; codes = atan([x,agg]@we+be); out = codes@wd+bd
__global__ void __launch_bounds__(256) k_out(const float* __restrict__ x,
                                             const float* __restrict__ we, const float* __restrict__ be,
                                             const float* __restrict__ wd, const float* __restrict__ bd,
                                             const float* __restrict__ ws, float* __restrict__ out) {
    int i = blockIdx.x * blockDim.x + threadIdx.x;
    const float* __restrict__ G = ws + WS_G;
    float inv = 1.0f / G[i * 8 + 7];

    float in14[14];
#pragma unroll
    for (int d = 0; d < 7; ++d) in14[d] = x[i * 7 + d];
#pragma unroll
    for (int d = 0; d < 7; ++d) in14[7 + d] = G[i * 8 + d] * inv;

    float codes[8];
#pragma unroll
    for (int h = 0; h < 8; ++h) {
        float acc = be[h];
#pragma unroll
        for (int d = 0; d < 14; ++d) acc += in14[d] * we[d * 8 + h];
        codes[h] = atanf(acc);
    }
#pragma unroll
    for (int d = 0; d < 7; ++d) {
        float acc = bd[d];
#pragma unroll
        for (int h = 0; h < 8; ++h) acc += codes[h] * wd[h * 7 + d];
        out[i * 7 + d] = acc;
    }
}

extern "C" void kernel_launch(void* const* d_in, const int* in_sizes, int n_in,
                              void* d_out, int out_size, void* d_ws, size_t ws_size,
                              hipStream_t stream) {
    (void)in_sizes; (void)n_in; (void)out_size; (void)ws_size;
    const float* x  = (const float*)d_in[0];
    const float* w1 = (const float*)d_in[1];
    const float* b1 = (const float*)d_in[2];
    const float* w2 = (const float*)d_in[3];
    const float* b2 = (const float*)d_in[4];
    const float* w3 = (const float*)d_in[5];
    const float* b3 = (const float*)d_in[6];
    const float* we = (const float*)d_in[7];
    const float* be = (const float*)d_in[8];
    const float* wd = (const float*)d_in[9];
    const float* bd = (const float*)d_in[10];
    float* ws  = (float*)d_ws;
    float* out = (float*)d_out;

    k_mlp   <<<NN / 256, 256, 0, stream>>>(x, w1, b1, w2, b2, w3, b3, ws);
    k_maxred<<<1, 32, 0, stream>>>(ws, NN / 256);
    k_buildF<<<NN / 256, 256, 0, stream>>>(ws);
    k_adjmm <<<NN / (16 * 8 * TPW), 256, 0, stream>>>(ws);   // 8 blocks x 8 waves x 4 tiles
    k_scan  <<<1, 32, 0, stream>>>(ws);
    k_out   <<<NN / 256, 256, 0, stream>>>(x, we, be, wd, bd, ws, out);
}